// BDH_GPU_34256659153645
// MI455X (gfx1250) — compile-verified
//
// MI455X / gfx1250 — BDH recurrent scan.
// Design: precompute relu(emb@Dx^T) with V_WMMA_F32_16X16X4_F32 (f32-exact),
// precompute ln(emb) rows; then ONE persistent kernel runs the T=128 sequential
// steps with software grid barriers. rho (16.8MB) stays L2-resident; the fused
// rho pass (dot + decay-update, 33.6MB L2 traffic/step) is the roofline term.
#include <hip/hip_runtime.h>

#define NN 2048
#define DD 256
#define BB 8
#define TT 128
#define NBLOCKS 128   // persistent grid: 128 blocks * 256 thr = 1024 waves

__constant__ const float kUDecay = 0.97f;
__constant__ const float kXDecay = 0.97f;

typedef __attribute__((ext_vector_type(2))) float v2f;
typedef __attribute__((ext_vector_type(8))) float v8f;

// ---- workspace layout (float offsets) ----
#define OFF_RHO   0u          // B*D*N = 4194304
#define OFF_XUPD  4194304u    // B*T*N = 2097152
#define OFF_VN    6291456u    // B*T*D = 262144
#define OFF_X     6553600u    // B*N   = 16384
#define OFF_AST   6569984u    // B*D   = 2048
#define OFF_Y     6572032u    // B*N   = 16384
#define OFF_VRAW  6588416u    // B*D   = 2048
#define OFF_BAR   6590464u    // 2 x u32 barrier state

__device__ __forceinline__ float wave_sum(float v) {
  for (int o = 16; o > 0; o >>= 1) v += __shfl_down(v, o, 32);
  return v;
}

// 256-thread block sum (8 waves of 32)
__device__ __forceinline__ float block_sum256(float v, float* sm) {
  v = wave_sum(v);
  if ((threadIdx.x & 31) == 0) sm[threadIdx.x >> 5] = v;
  __syncthreads();
  if (threadIdx.x < 32) {
    float w = (threadIdx.x < 8) ? sm[threadIdx.x] : 0.0f;
    for (int o = 4; o > 0; o >>= 1) w += __shfl_down(w, o, 32);
    if (threadIdx.x == 0) sm[0] = w;
  }
  __syncthreads();
  float r = sm[0];
  __syncthreads();
  return r;
}

__device__ __forceinline__ void grid_barrier(unsigned* bar, int nblocks) {
  __syncthreads();
  if (threadIdx.x == 0) {
    unsigned g = __hip_atomic_load(&bar[1], __ATOMIC_RELAXED, __HIP_MEMORY_SCOPE_AGENT);
    __threadfence();
    unsigned a = __hip_atomic_fetch_add(&bar[0], 1u, __ATOMIC_ACQ_REL, __HIP_MEMORY_SCOPE_AGENT);
    if (a == (unsigned)nblocks - 1u) {
      __hip_atomic_store(&bar[0], 0u, __ATOMIC_RELAXED, __HIP_MEMORY_SCOPE_AGENT);
      __hip_atomic_fetch_add(&bar[1], 1u, __ATOMIC_RELEASE, __HIP_MEMORY_SCOPE_AGENT);
    } else {
      while (__hip_atomic_load(&bar[1], __ATOMIC_ACQUIRE, __HIP_MEMORY_SCOPE_AGENT) == g)
        __builtin_amdgcn_s_sleep(2);
    }
    __threadfence();
  }
  __syncthreads();
}

// ---- init: copy state into workspace, zero barrier ----
__global__ void init_kernel(const float* __restrict__ rho_in,
                            const float* __restrict__ x_in, float* __restrict__ ws) {
  size_t i = (size_t)blockIdx.x * blockDim.x + threadIdx.x;
  size_t stride = (size_t)gridDim.x * blockDim.x;
  for (size_t k = i; k < (size_t)BB * DD * NN; k += stride) ws[OFF_RHO + k] = rho_in[k];
  for (size_t k = i; k < (size_t)BB * NN; k += stride) ws[OFF_X + k] = x_in[k];
  if (i == 0) { unsigned* b = (unsigned*)(ws + OFF_BAR); b[0] = 0u; b[1] = 0u; }
}

// ---- row layernorm over last dim D (one block per row, 256 threads) ----
__global__ void ln_rows_kernel(const float* __restrict__ src, float* __restrict__ dst) {
  __shared__ float sm[32];
  const int m = blockIdx.x, tid = threadIdx.x;
  float v = src[(size_t)m * DD + tid];
  float mean = block_sum256(v, sm) * (1.0f / DD);
  float d = v - mean;
  float var = block_sum256(d * d, sm) * (1.0f / DD);
  dst[(size_t)m * DD + tid] = d * rsqrtf(var + 1e-5f);
}

// ---- xupd = relu(emb(M=1024,K=256) @ Dx^T(K=256,N=2048)) via f32 WMMA ----
// A frag (16x4 f32): lane L holds M=L%16, K={k0+2*(L/16), +1} -> contiguous float2.
// B frag (4x16):    lane L holds N=L%16, K={k0+2*(L/16), +1} = Dx[n][k] float2.
__global__ void xupd_wmma_kernel(const float* __restrict__ emb,
                                 const float* __restrict__ dx,
                                 float* __restrict__ xupd) {
  const int lane = threadIdx.x & 31;
  const int wave = threadIdx.x >> 5;
  const int m0 = blockIdx.x * 16;
  const int n0 = blockIdx.y * 512 + wave * 64;   // 4 tiles of 16 columns
  const int koff = (lane >> 4) * 2;
  const float* arow = emb + (size_t)(m0 + (lane & 15)) * DD + koff;
  const float* brow = dx + (size_t)(n0 + (lane & 15)) * DD + koff;

  v8f acc0 = {}, acc1 = {}, acc2 = {}, acc3 = {};
  for (int k0 = 0; k0 < DD; k0 += 4) {
    v2f a  = *(const v2f*)(arow + k0);
    v2f b0 = *(const v2f*)(brow + k0);
    v2f b1 = *(const v2f*)(brow + 16 * DD + k0);
    v2f b2 = *(const v2f*)(brow + 32 * DD + k0);
    v2f b3 = *(const v2f*)(brow + 48 * DD + k0);
    acc0 = __builtin_amdgcn_wmma_f32_16x16x4_f32(false, a, false, b0, (short)0, acc0, false, false);
    acc1 = __builtin_amdgcn_wmma_f32_16x16x4_f32(false, a, false, b1, (short)0, acc1, false, false);
    acc2 = __builtin_amdgcn_wmma_f32_16x16x4_f32(false, a, false, b2, (short)0, acc2, false, false);
    acc3 = __builtin_amdgcn_wmma_f32_16x16x4_f32(false, a, false, b3, (short)0, acc3, false, false);
  }
  // C/D layout: VGPR j -> M = m0 + j + 8*(lane/16); N = n0 + lane%16
  const int ccol = lane & 15;
  const int mh = (lane >> 4) * 8;
  for (int j = 0; j < 8; ++j) {
    size_t row = (size_t)(m0 + mh + j) * NN + n0 + ccol;
    xupd[row +  0] = fmaxf(acc0[j], 0.0f);
    xupd[row + 16] = fmaxf(acc1[j], 0.0f);
    xupd[row + 32] = fmaxf(acc2[j], 0.0f);
    xupd[row + 48] = fmaxf(acc3[j], 0.0f);
  }
}

// ---- persistent scan over T steps ----
__global__ void scan_kernel(const float* __restrict__ dy, const float* __restrict__ e,
                            float* __restrict__ ws, float* __restrict__ out, int nblocks) {
  __shared__ float sm[32];
  __shared__ float lnA[BB * DD];   // 8 KB of 320 KB LDS
  float* rho   = ws + OFF_RHO;
  float* xupd  = ws + OFF_XUPD;
  float* vn    = ws + OFF_VN;
  float* x     = ws + OFF_X;
  float* astar = ws + OFF_AST;
  float* y     = ws + OFF_Y;
  float* vraw  = ws + OFF_VRAW;
  unsigned* bar = (unsigned*)(ws + OFF_BAR);
  const int tid = threadIdx.x, lane = tid & 31, wave = tid >> 5;
  const int gw = blockIdx.x * 8 + wave;               // global wave id (0..1023)
  const int gwn = nblocks * 8;

  for (int t = 0; t <= TT; ++t) {
    // Phase A: blocks 0..7 -> x-update(t); blocks 8..15 -> output LN of step t-1
    if (blockIdx.x < BB && t < TT) {
      const int b = blockIdx.x;
      const float* xu = xupd + (size_t)(b * TT + t) * NN;
      float raw[8], part = 0.0f;
      for (int j = 0; j < 8; ++j) {
        int n = j * 256 + tid;
        raw[j] = kXDecay * x[b * NN + n] + xu[n];
        part += fabsf(raw[j]);
      }
      float s = block_sum256(part, sm);
      float inv = 1.0f / fmaxf(s, 1e-12f);
      for (int j = 0; j < 8; ++j) { int n = j * 256 + tid; x[b * NN + n] = raw[j] * inv; }
    } else if (blockIdx.x >= BB && blockIdx.x < 2 * BB && t > 0) {
      const int b = blockIdx.x - BB;
      float v = vraw[b * DD + tid];
      float mean = block_sum256(v, sm) * (1.0f / DD);
      float d = v - mean;
      float var = block_sum256(d * d, sm) * (1.0f / DD);
      out[((size_t)b * TT + (t - 1)) * DD + tid] = d * rsqrtf(var + 1e-5f);
    }
    if (t == TT) break;
    grid_barrier(bar, nblocks);

    // Phase R: fused a_star = rho . x_t  AND  rho = 0.97*rho + ln(emb_t) (x) x_t
    for (int r = blockIdx.x; r < BB * DD; r += nblocks) {
      const int b = r >> 8, d = r & (DD - 1);
      const float vnd = vn[(size_t)(b * TT + t) * DD + d];
      float* rp = rho + (size_t)r * NN + tid * 8;
      const float* xp = x + (size_t)b * NN + tid * 8;
      float4 r0 = *(float4*)(rp), r1 = *(float4*)(rp + 4);
      float4 x0 = *(const float4*)(xp), x1 = *(const float4*)(xp + 4);
      float acc = r0.x * x0.x + r0.y * x0.y + r0.z * x0.z + r0.w * x0.w
                + r1.x * x1.x + r1.y * x1.y + r1.z * x1.z + r1.w * x1.w;
      r0.x = kUDecay * r0.x + vnd * x0.x;  r0.y = kUDecay * r0.y + vnd * x0.y;
      r0.z = kUDecay * r0.z + vnd * x0.z;  r0.w = kUDecay * r0.w + vnd * x0.w;
      r1.x = kUDecay * r1.x + vnd * x1.x;  r1.y = kUDecay * r1.y + vnd * x1.y;
      r1.z = kUDecay * r1.z + vnd * x1.z;  r1.w = kUDecay * r1.w + vnd * x1.w;
      *(float4*)(rp) = r0; *(float4*)(rp + 4) = r1;
      float s = block_sum256(acc, sm);
      if (tid == 0) astar[r] = s;
    }
    grid_barrier(bar, nblocks);

    // Phase LA+Y: every block computes ln(a_star) redundantly into LDS, then
    // y[b,n] = relu(lnA . Dy[n,:]) * relu(x[b,n]) (one output per wave-iter)
    for (int b = 0; b < BB; ++b) {
      float v = astar[b * DD + tid];
      float mean = block_sum256(v, sm) * (1.0f / DD);
      float d = v - mean;
      float var = block_sum256(d * d, sm) * (1.0f / DD);
      lnA[b * DD + tid] = d * rsqrtf(var + 1e-5f);
    }
    __syncthreads();
    for (int idx = gw; idx < BB * NN; idx += gwn) {
      const int b = idx >> 11, n = idx & (NN - 1);
      const float* dyr = dy + (size_t)n * DD;
      float acc = 0.0f;
      for (int j = 0; j < 8; ++j) { int k = lane + 32 * j; acc += lnA[b * DD + k] * dyr[k]; }
      acc = wave_sum(acc);
      if (lane == 0) y[idx] = fmaxf(acc, 0.0f) * fmaxf(x[idx], 0.0f);
    }
    grid_barrier(bar, nblocks);

    // Phase V: vraw[b,d] = y[b,:] . E[d,:]  (K = 2048 per wave)
    for (int idx = gw; idx < BB * DD; idx += gwn) {
      const int b = idx >> 8, d = idx & (DD - 1);
      const float* er = e + (size_t)d * NN;
      const float* yb = y + (size_t)b * NN;
      float acc = 0.0f;
      for (int j = 0; j < 64; ++j) { int k = lane + 32 * j; acc += yb[k] * er[k]; }
      acc = wave_sum(acc);
      if (lane == 0) vraw[idx] = acc;
    }
    grid_barrier(bar, nblocks);
  }
}

extern "C" void kernel_launch(void* const* d_in, const int* in_sizes, int n_in,
                              void* d_out, int out_size, void* d_ws, size_t ws_size,
                              hipStream_t stream) {
  const float* emb  = (const float*)d_in[0];   // (B,T,D)
  const float* E    = (const float*)d_in[1];   // (D,N)
  const float* Dx   = (const float*)d_in[2];   // (N,D)
  const float* Dy   = (const float*)d_in[3];   // (N,D)
  const float* x0   = (const float*)d_in[4];   // (B,N)
  const float* rho0 = (const float*)d_in[5];   // (B,D,N)
  float* out = (float*)d_out;
  float* ws  = (float*)d_ws;
  (void)in_sizes; (void)n_in; (void)out_size; (void)ws_size;

  init_kernel<<<2048, 256, 0, stream>>>(rho0, x0, ws);
  ln_rows_kernel<<<BB * TT, 256, 0, stream>>>(emb, ws + OFF_VN);
  xupd_wmma_kernel<<<dim3((BB * TT) / 16, NN / 512), 256, 0, stream>>>(emb, Dx, ws + OFF_XUPD);
  scan_kernel<<<NBLOCKS, 256, 0, stream>>>(Dy, E, ws, out, NBLOCKS);
}